// CRF_37357625541027
// MI455X (gfx1250) — compile-verified
//
#include <hip/hip_runtime.h>
#include <cstdint>
#include <cstddef>

typedef __attribute__((ext_vector_type(2))) float v2f;
typedef __attribute__((ext_vector_type(4))) float v4f;
typedef __attribute__((ext_vector_type(8))) float v8f;

#define CRF_L 512
#define CRF_B 1024
#define CRF_T 48
#define EROW 50   // padded LDS row stride (floats); even -> 8B-aligned b64, conflict-free banks

// ---------------------------------------------------------------------------
// Zero the scalar output (harness poisons d_out; atomics accumulate into it).
// ---------------------------------------------------------------------------
__global__ void crf_zero_out(float* out) {
    if (threadIdx.x == 0 && blockIdx.x == 0) out[0] = 0.0f;
}

// ---------------------------------------------------------------------------
// Permuted exp-transition matrix.
//   M-space (rows p, C/D layout): identity tag order.
//   K-space (cols q, B layout):  tag order tau chosen so each tag's K-position
//   lands in the same lane half as its M-position:
//     q = 4*kt + c:  c in {0,1} -> lo[2kt+c],  c in {2,3} -> hi[2kt+c-2]
//     lo[j] = 16*(j/8) + j%8,  hi[j] = lo[j] + 8
//   ET2[p*48 + q] = exp(trans[tau_inv(q)][p])
// This makes next-step B-fragments pure per-lane register reads (no shuffles).
// ---------------------------------------------------------------------------
__global__ void crf_prep_ET(const float* __restrict__ trans, float* __restrict__ ET2) {
    int idx = blockIdx.x * blockDim.x + threadIdx.x;
    if (idx < CRF_T * CRF_T) {
        int p  = idx / CRF_T;
        int q  = idx % CRF_T;
        int kt = q >> 2;
        int c  = q & 3;
        int j  = 2 * kt + (c & 1);
        int i  = 16 * (j >> 3) + (j & 7) + ((c >= 2) ? 8 : 0);   // tau_inv(q)
        ET2[p * CRF_T + q] = __expf(trans[i * CRF_T + p]);
    }
}

// ---------------------------------------------------------------------------
// Numerator: one thread per batch, exact reference formulas, atomicAdd(+llh).
// ---------------------------------------------------------------------------
__global__ void crf_numerator(const float* __restrict__ emis,
                              const int*   __restrict__ tags,
                              const int*   __restrict__ mask,
                              const float* __restrict__ start_t,
                              const float* __restrict__ end_t,
                              const float* __restrict__ trans,
                              float* out) {
    int b = blockIdx.x * blockDim.x + threadIdx.x;
    if (b >= CRF_B) return;

    int tag0 = tags[b];
    float llh = start_t[tag0];
    int msum = 0;
    int prev = tag0;
    for (int t = 0; t < CRF_L; ++t) {
        int tg = (t == 0) ? tag0 : tags[t * CRF_B + b];
        int mk = mask[t * CRF_B + b];
        msum += mk;
        if (t < CRF_L - 1 && mk > 0)
            llh += emis[((size_t)t * CRF_B + b) * CRF_T + tg];   // emis-at-tag, mask[t]
        if (t >= 1 && mk > 0)
            llh += trans[prev * CRF_T + tg];                     // pair (t-1,t), mask[t]
        prev = tg;
    }
    int last_idx = msum - 1;
    int last_tag = tags[(size_t)last_idx * CRF_B + b];
    llh += end_t[last_tag];
    if (mask[(CRF_L - 1) * CRF_B + b] > 0)
        llh += emis[((size_t)(CRF_L - 1) * CRF_B + b) * CRF_T + last_tag];

    atomicAdd(out, llh);
}

// ---------------------------------------------------------------------------
// Forward algorithm. One wave owns 16 batch columns for all 512 steps.
// State lp kept in C/D WMMA layout (tags x batch): acc[mt][r] = lp[tag][b],
// tag = 16*mt + r + (lane<16 ? 0 : 8), b = b0 + (lane&15).  Per step:
//   D(tags x batch) = ET2(48x48, K-permuted) @ P~(48 x 16),  P = exp(lp - m)
//   lp' = mask ? (m + log(D) + emis[t]) : lp
// Thanks to the K-permutation baked into ET2, B-fragments are direct per-lane
// register reads of acc — the only cross-lane op per step is one xor-16 max.
// K-reduction uses two independent accumulator banks (6-deep WMMA chains).
// ---------------------------------------------------------------------------
__launch_bounds__(128, 1)
__global__ void crf_forward(const float* __restrict__ emis,
                            const int*   __restrict__ mask,
                            const float* __restrict__ start_t,
                            const float* __restrict__ end_t,
                            const float* __restrict__ ET2,  // [T*T] permuted exp(trans)
                            float* out) {
    __shared__ float sE[CRF_T * EROW];

    // Stage ET2 into LDS with padded rows (row p = output tag, col q = K-pos)
    for (int idx = threadIdx.x; idx < CRF_T * CRF_T; idx += blockDim.x) {
        int p = idx / CRF_T, q = idx % CRF_T;
        sE[p * EROW + q] = ET2[idx];
    }
    __syncthreads();

    const int lane = threadIdx.x & 31;
    const int wave = threadIdx.x >> 5;
    const int col  = lane & 15;          // batch column within the 16-wide tile
    const bool lo  = lane < 16;          // low lane-half flag
    const int hofs = lo ? 0 : 8;         // this half's tag sub-offset
    const int b    = blockIdx.x * 64 + wave * 16 + col;
    const int Msub = lane & 15;          // M row within a 16-tag tile (A operand)
    const int kofs = lo ? 0 : 2;         // this half's K sub-offset (A/B operands)

    // ---- init: lp = start + emissions[0] (vectorized: 8 consecutive tags/lane) ----
    v8f acc[3];
    {
        const float* erow = emis + (size_t)b * CRF_T + hofs;
        #pragma unroll
        for (int mt = 0; mt < 3; ++mt) {
            v4f ea = *(const v4f*)(erow + 16 * mt);
            v4f eb = *(const v4f*)(erow + 16 * mt + 4);
            v4f sa = *(const v4f*)(start_t + 16 * mt + hofs);
            v4f sb = *(const v4f*)(start_t + 16 * mt + hofs + 4);
            #pragma unroll
            for (int r = 0; r < 4; ++r) {
                acc[mt][r]     = sa[r] + ea[r];
                acc[mt][r + 4] = sb[r] + eb[r];
            }
        }
    }

    // ---- recurrence over t = 1..L-1 ----
    for (int t = 1; t < CRF_L; ++t) {
        // per-batch max: local max over this half's 24 tags, combine halves (1 shfl)
        float m = acc[0][0];
        #pragma unroll
        for (int mt = 0; mt < 3; ++mt)
            #pragma unroll
            for (int r = 0; r < 8; ++r) m = fmaxf(m, acc[mt][r]);
        m = fmaxf(m, __shfl_xor(m, 16, 32));

        if (t + 1 < CRF_L)
            __builtin_prefetch(&emis[((size_t)(t + 1) * CRF_B + b) * CRF_T], 0, 3);

        // emissions for this step: 6 x b128 (lane's 8 consecutive tags per tile)
        const float* erow = emis + ((size_t)t * CRF_B + b) * CRF_T + hofs;
        v4f e[6];
        #pragma unroll
        for (int mt = 0; mt < 3; ++mt) {
            e[2 * mt]     = *(const v4f*)(erow + 16 * mt);
            e[2 * mt + 1] = *(const v4f*)(erow + 16 * mt + 4);
        }
        int mk = mask[t * CRF_B + b];

        // D = ET2 @ P~ : B-fragments are direct register reads (K-permutation),
        // two accumulator banks halve the WMMA dependency chain.
        v8f dA[3], dB[3];
        #pragma unroll
        for (int ot = 0; ot < 3; ++ot) {
            dA[ot] = (v8f){0.f, 0.f, 0.f, 0.f, 0.f, 0.f, 0.f, 0.f};
            dB[ot] = (v8f){0.f, 0.f, 0.f, 0.f, 0.f, 0.f, 0.f, 0.f};
        }

        #pragma unroll
        for (int kt = 0; kt < 12; ++kt) {
            const int q0 = 2 * kt;          // acc element feeding bf.x (both halves)
            const int q1 = 2 * kt + 1;      // acc element feeding bf.y (both halves)
            v2f bf;
            bf.x = __expf(acc[q0 >> 3][q0 & 7] - m);
            bf.y = __expf(acc[q1 >> 3][q1 & 7] - m);
            #pragma unroll
            for (int ot = 0; ot < 3; ++ot) {
                v2f a = *(const v2f*)(&sE[(16 * ot + Msub) * EROW + 4 * kt + kofs]);
                if (kt & 1)
                    dB[ot] = __builtin_amdgcn_wmma_f32_16x16x4_f32(
                                false, a, false, bf, (short)0, dB[ot], false, false);
                else
                    dA[ot] = __builtin_amdgcn_wmma_f32_16x16x4_f32(
                                false, a, false, bf, (short)0, dA[ot], false, false);
            }
        }

        // lp' = mask ? m + log(dA + dB) + emis[t] : lp
        #pragma unroll
        for (int mt = 0; mt < 3; ++mt) {
            #pragma unroll
            for (int r = 0; r < 8; ++r) {
                float ev = e[2 * mt + (r >> 2)][r & 3];
                float ns = m + __logf(dA[mt][r] + dB[mt][r]) + ev;
                acc[mt][r] = (mk > 0) ? ns : acc[mt][r];
            }
        }
    }

    // ---- log_z = logsumexp(lp + end_transitions), 2 shuffles total ----
    float tmp[24];
    #pragma unroll
    for (int mt = 0; mt < 3; ++mt) {
        v4f za = *(const v4f*)(end_t + 16 * mt + hofs);
        v4f zb = *(const v4f*)(end_t + 16 * mt + hofs + 4);
        #pragma unroll
        for (int r = 0; r < 4; ++r) {
            tmp[mt * 8 + r]     = acc[mt][r]     + za[r];
            tmp[mt * 8 + r + 4] = acc[mt][r + 4] + zb[r];
        }
    }
    float m = tmp[0];
    #pragma unroll
    for (int k = 1; k < 24; ++k) m = fmaxf(m, tmp[k]);
    m = fmaxf(m, __shfl_xor(m, 16, 32));
    float s = 0.f;
    #pragma unroll
    for (int k = 0; k < 24; ++k) s += __expf(tmp[k] - m);
    s += __shfl_xor(s, 16, 32);
    float logz = m + __logf(s);

    if (lo) atomicAdd(out, -logz);   // one add per batch column
}

// ---------------------------------------------------------------------------
// launch
// ---------------------------------------------------------------------------
extern "C" void kernel_launch(void* const* d_in, const int* in_sizes, int n_in,
                              void* d_out, int out_size, void* d_ws, size_t ws_size,
                              hipStream_t stream) {
    const float* emissions = (const float*)d_in[0];   // [L,B,T]
    const int*   tags      = (const int*)  d_in[1];   // [L,B]
    const int*   mask      = (const int*)  d_in[2];   // [L,B]
    const float* start_t   = (const float*)d_in[3];   // [T]
    const float* end_t     = (const float*)d_in[4];   // [T]
    const float* trans     = (const float*)d_in[5];   // [T,T]
    float* out = (float*)d_out;                       // scalar
    float* ET2 = (float*)d_ws;                        // [T*T] permuted exp(trans)

    crf_zero_out<<<1, 32, 0, stream>>>(out);
    crf_prep_ET<<<(CRF_T * CRF_T + 255) / 256, 256, 0, stream>>>(trans, ET2);
    crf_numerator<<<CRF_B / 256, 256, 0, stream>>>(emissions, tags, mask,
                                                   start_t, end_t, trans, out);
    // 16 workgroups x 4 waves x 16 batch columns = 1024 batches
    crf_forward<<<CRF_B / 64, 128, 0, stream>>>(emissions, mask, start_t, end_t, ET2, out);
}